// DeeProBot_MoE_gate_52518860095673
// MI455X (gfx1250) — compile-verified
//
#include <hip/hip_runtime.h>

typedef float v2f __attribute__((ext_vector_type(2)));
typedef float v8f __attribute__((ext_vector_type(8)));

#define NN    11826
#define DIN   9
#define DGATE 36
#define NE    2
#define HH    128
#define MO    32
#define NO    2

__global__ void moe_zero_ws(int* cnt) {
    if (threadIdx.x == 0) cnt[0] = 0;
}

__global__ __launch_bounds__(32) void moe_tile_kernel(
    const float* __restrict__ x,      // (N, DIN)
    const float* __restrict__ gate,   // (N, DGATE)
    const float* __restrict__ w_gate, // (DGATE, E)
    const float* __restrict__ w1,     // (E, DIN, H)
    const float* __restrict__ b1,     // (E, H)
    const float* __restrict__ w2,     // (E, H, MO)
    const float* __restrict__ b2,     // (E, MO)
    const float* __restrict__ wo,     // (MO, NO)
    const float* __restrict__ bo,     // (NO,)
    float* __restrict__ out,          // (N, NO)
    int* __restrict__ cnt1)           // expert-1 row count
{
    __shared__ float h_lds[16 * HH];        // 8 KB: h tile, reused per expert
    __shared__ float y_lds[NE * 16 * MO];   // 4 KB: y for both experts
    __shared__ int   sel[16];

    const int lane = threadIdx.x;           // wave32
    const int row0 = blockIdx.x * 16;
    const int m    = lane & 15;             // row-in-tile (A "M") / col ("N")
    const int hi   = lane >> 4;             // half-wave select
    const int rowm = min(row0 + m, NN - 1); // clamp: all loads in-bounds

    // ---------------- gating: lanes 0..15, one row each ----------------
    if (lane < 16) {
        const float* g = gate + (size_t)rowm * DGATE;
        float l0 = 0.f, l1 = 0.f;
        #pragma unroll
        for (int d = 0; d < DGATE; ++d) {
            float gv = g[d];
            l0 += gv * w_gate[d * NE + 0];
            l1 += gv * w_gate[d * NE + 1];
        }
        // top_k picks lower index on ties -> expert 1 only if strictly greater
        int e = (l1 > l0) ? 1 : 0;
        sel[lane] = e;
        if ((row0 + lane) < NN && e == 1) atomicAdd(cnt1, 1);
    }
    __syncthreads();

    // ---- x tile in WMMA A layout (16 x K, K=9 zero-padded to 12) ----
    // Branch-free padding: clamp the address (always valid), cndmask the value.
    v2f ax[3];
    {
        const float* xr = x + (size_t)rowm * DIN;
        #pragma unroll
        for (int c = 0; c < 3; ++c) {
            const int k0 = c * 4 + hi * 2;
            float v0 = xr[min(k0,     DIN - 1)];   // unconditional load
            float v1 = xr[min(k0 + 1, DIN - 1)];   // unconditional load
            v2f a;
            a.x = (k0     < DIN) ? v0 : 0.f;       // v_cndmask, no exec change
            a.y = (k0 + 1 < DIN) ? v1 : 0.f;
            ax[c] = a;
        }
    }

    for (int e = 0; e < NE; ++e) {
        const float* W1 = w1 + (size_t)e * DIN * HH;
        const float* W2 = w2 + (size_t)e * HH * MO;
        __builtin_prefetch(W2, 0, 1);              // global_prefetch_b8

        // ------- h = relu(x @ W1 + b1): 8 column tiles of 16 -------
        for (int nt = 0; nt < 8; ++nt) {
            const int n0 = nt * 16;
            const float bias = b1[e * HH + n0 + m];  // fixed column per lane
            v8f c8;
            #pragma unroll
            for (int i = 0; i < 8; ++i) c8[i] = bias;
            #pragma unroll
            for (int cix = 0; cix < 3; ++cix) {
                const int kb = cix * 4 + hi * 2;     // B rows K=kb..kb+1
                float w0 = W1[min(kb,     DIN - 1) * HH + n0 + m];
                float w1v = W1[min(kb + 1, DIN - 1) * HH + n0 + m];
                v2f b;
                b.x = (kb     < DIN) ? w0  : 0.f;
                b.y = (kb + 1 < DIN) ? w1v : 0.f;
                c8 = __builtin_amdgcn_wmma_f32_16x16x4_f32(
                         false, ax[cix], false, b, (short)0, c8, false, false);
            }
            // relu + scatter C layout to LDS: VGPR i -> row (i + hi*8), col n0+m
            #pragma unroll
            for (int i = 0; i < 8; ++i)
                h_lds[(i + hi * 8) * HH + n0 + m] = fmaxf(c8[i], 0.0f);
        }
        __syncthreads();

        // ------- y = h @ W2 + b2: 2 column tiles of 16, K=128 -------
        for (int nt = 0; nt < 2; ++nt) {
            const int n0 = nt * 16;
            const float bias = b2[e * MO + n0 + m];
            v8f c8;
            #pragma unroll
            for (int i = 0; i < 8; ++i) c8[i] = bias;
            #pragma unroll 4
            for (int k = 0; k < HH; k += 4) {
                const int ka = k + hi * 2;           // even -> 8B aligned
                v2f a = *(const v2f*)&h_lds[m * HH + ka];  // ds_load_b64
                v2f b;
                b.x = W2[(ka    ) * MO + n0 + m];
                b.y = W2[(ka + 1) * MO + n0 + m];
                c8 = __builtin_amdgcn_wmma_f32_16x16x4_f32(
                         false, a, false, b, (short)0, c8, false, false);
            }
            #pragma unroll
            for (int i = 0; i < 8; ++i)
                y_lds[e * (16 * MO) + (i + hi * 8) * MO + n0 + m] = c8[i];
        }
        __syncthreads();
    }

    // -------- final projection with per-row expert select --------
    if (lane < 16 && (row0 + lane) < NN) {
        const int e = sel[lane];
        const float* yr = y_lds + e * (16 * MO) + lane * MO;
        float a0 = bo[0], a1 = bo[1];
        #pragma unroll
        for (int j = 0; j < MO; ++j) {
            float yv = yr[j];
            a0 += yv * wo[j * NO + 0];
            a1 += yv * wo[j * NO + 1];
        }
        out[(row0 + lane) * NO + 0] = a0;
        out[(row0 + lane) * NO + 1] = a1;
    }
}

__global__ void moe_loss_kernel(const int* __restrict__ cnt1,
                                float* __restrict__ loss_out) {
    if (threadIdx.x != 0) return;
    const float c1 = (float)cnt1[0];
    const float c0 = (float)NN - c1;
    const float mv = 0.5f * (float)NN;
    // ddof=1 variance of a 2-vector: ((c0-m)^2 + (c1-m)^2) / 1
    const float var = (c0 - mv) * (c0 - mv) + (c1 - mv) * (c1 - mv);
    const float cv  = var / (mv * mv + 1e-10f);
    loss_out[0] = 2.0f * cv * 0.01f;   // importance and load are identical
}

extern "C" void kernel_launch(void* const* d_in, const int* in_sizes, int n_in,
                              void* d_out, int out_size, void* d_ws, size_t ws_size,
                              hipStream_t stream) {
    const float* x      = (const float*)d_in[0];
    const float* gate   = (const float*)d_in[1];
    const float* w_gate = (const float*)d_in[2];
    const float* w1     = (const float*)d_in[3];
    const float* b1     = (const float*)d_in[4];
    const float* w2     = (const float*)d_in[5];
    const float* b2     = (const float*)d_in[6];
    const float* wo     = (const float*)d_in[7];
    const float* bo     = (const float*)d_in[8];
    float* out = (float*)d_out;
    int*   cnt = (int*)d_ws;

    const int tiles = (NN + 15) / 16;  // 740

    moe_zero_ws<<<1, 1, 0, stream>>>(cnt);
    moe_tile_kernel<<<tiles, 32, 0, stream>>>(x, gate, w_gate, w1, b1, w2, b2,
                                              wo, bo, out, cnt);
    moe_loss_kernel<<<1, 1, 0, stream>>>(cnt, out + (size_t)NN * NO);
}